// Norm1d_24790551233315
// MI455X (gfx1250) — compile-verified
//
#include <hip/hip_runtime.h>
#include <hip/hip_bf16.h>

// Online Normalization forward for MI455X (gfx1250).
//
// HBM-bound: 256 MiB minimum traffic -> ~11 us @ 23.3 TB/s. The serial EMA
// over N is split into K chunks via an exact affine/quadratic transfer-
// function scan (4096 independent column-chunk waves). Data movement uses the
// CDNA5 async path: GLOBAL_LOAD_ASYNC_TO_LDS_B128 double-buffered stages,
// flow-controlled with s_wait_asynccnt (in-order completion => exact waits,
// lane-private LDS slots => no barriers), plus TH=NT output stores so the
// 128 MB out-stream does not evict x from the 192 MB L2 between passes.
// No WMMA: there is no matmul in this recurrence (2 FMAs/element of scan).

typedef __attribute__((ext_vector_type(4))) float v4f;

constexpr int   NN   = 8192;
constexpr int   LL   = 4096;
constexpr float AF   = 0.999f;
constexpr float BF   = 1.0f - AF;      // 1 - afwd
constexpr float EPSF = 1e-5f;

constexpr int K  = 128;                 // chunks along N
constexpr int CH = NN / K;              // 64 rows per chunk
constexpr int SR = 8;                   // rows per async stage
constexpr int NSTAGE = CH / SR;         // 8 stages per chunk

constexpr float pow_ce(double base, int e) {
    double r = 1.0;
    for (int i = 0; i < e; ++i) r *= base;
    return (float)r;
}
constexpr float A_CH  = pow_ce((double)AF, CH);    // a^CH
constexpr float C2_CH = A_CH * (1.0f - A_CH);      // A(1-A), column-independent

__device__ __forceinline__ float rsq_nr(float v) {
    // fast v_rsq_f32 + one Newton step (VALU has slack; memory-bound kernel)
    float r = __builtin_amdgcn_rsqf(v);
    return r * (1.5f - 0.5f * v * r * r);
}

// ---- CDNA5 async copy: global -> LDS, 16B per lane, tracked on ASYNCcnt ----
__device__ __forceinline__ void async_row_b128(unsigned lds_addr, unsigned goff,
                                               const float* base) {
    // GVS mode: addr = SGPR64(base) + VGPR32(goff); vdst = per-lane LDS byte addr
    asm volatile("global_load_async_to_lds_b128 %0, %1, %2"
                 :
                 : "v"(lds_addr), "v"(goff), "s"(base)
                 : "memory");
}

__device__ __forceinline__ void wait_async_le(int n) {
#if __has_builtin(__builtin_amdgcn_s_wait_asynccnt)
    if (n == 0) __builtin_amdgcn_s_wait_asynccnt(0);
    else        __builtin_amdgcn_s_wait_asynccnt(SR);
#else
    if (n == 0) asm volatile("s_wait_asynccnt 0x0" ::: "memory");
    else        asm volatile("s_wait_asynccnt 0x8" ::: "memory");
#endif
}

// ---------------- Pass 1: per-chunk local transfer coefficients ------------
// For chunk c, column j:  m (local EMA from 0), S1 = sum p_i,
// S2 = sum a^(n-i) p_i^2, with p_i = x_i - m_{i-1}.
__global__ void __launch_bounds__(256)
onorm_pass1(const float* __restrict__ x,
            float* __restrict__ M, float* __restrict__ S1, float* __restrict__ S2) {
    __shared__ float lds[2][SR][1024];          // 64 KB, lane-private 16B slots

    const int tid   = threadIdx.x;
    const int chunk = blockIdx.y;
    const int col   = blockIdx.x * 1024 + tid * 4;

    const unsigned g0 = (unsigned)(((size_t)chunk * CH * LL + col) * 4);
    const unsigned l0 = (unsigned)(uintptr_t)&lds[0][0][0] + (unsigned)tid * 16u;

    // prime two stages (16 async b128 in flight per wave)
#pragma unroll
    for (int s = 0; s < 2; ++s)
#pragma unroll
        for (int r = 0; r < SR; ++r)
            async_row_b128(l0 + (unsigned)(s * SR + r) * 4096u,
                           g0 + (unsigned)((s * SR + r) * LL) * 4u, x);

    v4f m  = {0.f, 0.f, 0.f, 0.f};
    v4f s1 = {0.f, 0.f, 0.f, 0.f};
    v4f s2 = {0.f, 0.f, 0.f, 0.f};

    for (int s = 0; s < NSTAGE; ++s) {
        wait_async_le((s + 1 < NSTAGE) ? SR : 0);   // oldest stage landed
        const int buf = s & 1;
#pragma unroll
        for (int r = 0; r < SR; ++r) {
            v4f xv = *(const v4f*)&lds[buf][r][tid * 4];
            v4f p  = xv - m;
            s1 = s1 + p;
            s2 = AF * s2 + p * p;                   // running sum a^(n-i) p^2
            m  = m + BF * p;                        // m <- a*m + b*x
        }
        if (s + 2 < NSTAGE) {                       // refill consumed buffer
            const unsigned ls = l0 + (unsigned)buf * (SR * 4096u);
            const unsigned gs = g0 + (unsigned)((s + 2) * SR * LL) * 4u;
#pragma unroll
            for (int r = 0; r < SR; ++r)
                async_row_b128(ls + (unsigned)r * 4096u,
                               gs + (unsigned)(r * LL) * 4u, x);
        }
    }

    const size_t o = (size_t)chunk * LL + col;
    *(v4f*)(M  + o) = m;
    *(v4f*)(S1 + o) = s1;
    *(v4f*)(S2 + o) = s2;
}

// ---------------- Pass 2: sequential chunk combine (tiny) ------------------
// mu_{c+1}  = A*mu_c + M_c
// var_{c+1} = A*var_c + a*b*S2_c - 2*b*A*S1_c*mu_c + A(1-A)*mu_c^2
__global__ void __launch_bounds__(256)
onorm_pass2(const float* __restrict__ mu0, const float* __restrict__ var0,
            const float* __restrict__ M, const float* __restrict__ S1,
            const float* __restrict__ S2,
            float* __restrict__ MuS, float* __restrict__ VarS) {
    const int col = (blockIdx.x * blockDim.x + threadIdx.x) * 4;
    v4f mu  = *(const v4f*)(mu0 + col);
    v4f var = *(const v4f*)(var0 + col);

    for (int c = 0; c < K; ++c) {
        const size_t o = (size_t)c * LL + col;
        *(v4f*)(MuS + o)  = mu;                  // state at START of chunk c
        *(v4f*)(VarS + o) = var;
        v4f Mv  = *(const v4f*)(M  + o);
        v4f S1v = *(const v4f*)(S1 + o);
        v4f S2v = *(const v4f*)(S2 + o);
        v4f C0 = (AF * BF) * S2v;
        v4f C1 = (2.0f * BF * A_CH) * S1v;
        var = A_CH * var + C0 - C1 * mu + C2_CH * (mu * mu);
        mu  = A_CH * mu + Mv;
    }
}

// ---------------- Pass 3: exact reference recurrence per chunk -------------
__global__ void __launch_bounds__(256)
onorm_pass3(const float* __restrict__ x,
            const float* __restrict__ MuS, const float* __restrict__ VarS,
            float* __restrict__ out) {
    __shared__ float lds[2][SR][1024];

    const int tid   = threadIdx.x;
    const int chunk = blockIdx.y;
    const int col   = blockIdx.x * 1024 + tid * 4;

    const unsigned g0 = (unsigned)(((size_t)chunk * CH * LL + col) * 4);
    const unsigned l0 = (unsigned)(uintptr_t)&lds[0][0][0] + (unsigned)tid * 16u;

#pragma unroll
    for (int s = 0; s < 2; ++s)
#pragma unroll
        for (int r = 0; r < SR; ++r)
            async_row_b128(l0 + (unsigned)(s * SR + r) * 4096u,
                           g0 + (unsigned)((s * SR + r) * LL) * 4u, x);

    const size_t so = (size_t)chunk * LL + col;
    v4f mu  = *(const v4f*)(MuS + so);
    v4f var = *(const v4f*)(VarS + so);

    float* op = out + (size_t)chunk * CH * LL + col;

    for (int s = 0; s < NSTAGE; ++s) {
        wait_async_le((s + 1 < NSTAGE) ? SR : 0);
        const int buf = s & 1;
#pragma unroll
        for (int r = 0; r < SR; ++r) {
            v4f xv = *(const v4f*)&lds[buf][r][tid * 4];
            v4f d  = xv - mu;
            v4f sc = var + EPSF;
            v4f rc;
            rc.x = rsq_nr(sc.x); rc.y = rsq_nr(sc.y);
            rc.z = rsq_nr(sc.z); rc.w = rsq_nr(sc.w);
            // NT store: out-stream must not evict x from L2.
            __builtin_nontemporal_store(d * rc, (v4f*)op);
            var = AF * var + (AF * BF) * (d * d);
            mu  = mu + BF * d;
            op += LL;
        }
        if (s + 2 < NSTAGE) {
            const unsigned ls = l0 + (unsigned)buf * (SR * 4096u);
            const unsigned gs = g0 + (unsigned)((s + 2) * SR * LL) * 4u;
#pragma unroll
            for (int r = 0; r < SR; ++r)
                async_row_b128(ls + (unsigned)r * 4096u,
                               gs + (unsigned)(r * LL) * 4u, x);
        }
    }
}

// ---------------- Fallback: single pass, used only if ws too small ---------
__global__ void __launch_bounds__(256)
onorm_fallback(const float* __restrict__ x, const float* __restrict__ mu0,
               const float* __restrict__ var0, float* __restrict__ out) {
    const int col = (blockIdx.x * blockDim.x + threadIdx.x) * 4;
    v4f mu  = *(const v4f*)(mu0 + col);
    v4f var = *(const v4f*)(var0 + col);
    const float* xp = x + col;
    float*       op = out + col;
#pragma unroll 8
    for (int t = 0; t < NN; ++t) {
        v4f xv = __builtin_nontemporal_load((const v4f*)xp);
        __builtin_prefetch(xp + 16 * LL, 0, 0);   // global_prefetch_b8
        v4f d = xv - mu;
        v4f s = var + EPSF;
        v4f r;
        r.x = rsq_nr(s.x); r.y = rsq_nr(s.y); r.z = rsq_nr(s.z); r.w = rsq_nr(s.w);
        __builtin_nontemporal_store(d * r, (v4f*)op);
        var = AF * var + (AF * BF) * (d * d);
        mu  = mu + BF * d;
        xp += LL; op += LL;
    }
}

extern "C" void kernel_launch(void* const* d_in, const int* in_sizes, int n_in,
                              void* d_out, int out_size, void* d_ws, size_t ws_size,
                              hipStream_t stream) {
    (void)in_sizes; (void)n_in; (void)out_size;
    const float* x    = (const float*)d_in[0];
    const float* mu0  = (const float*)d_in[1];
    const float* var0 = (const float*)d_in[2];
    float*       out  = (float*)d_out;

    const size_t KL = (size_t)K * LL;
    const size_t needed = 5 * KL * sizeof(float);   // M,S1,S2,MuS,VarS = 10 MB

    if (ws_size >= needed) {
        float* w    = (float*)d_ws;
        float* M    = w;
        float* S1   = w + KL;
        float* S2   = w + 2 * KL;
        float* MuS  = w + 3 * KL;
        float* VarS = w + 4 * KL;

        dim3 blk(256);
        dim3 grd(LL / 1024, K);         // (4, 128) -> 512 blocks = 4096 waves
        onorm_pass1<<<grd, blk, 0, stream>>>(x, M, S1, S2);
        onorm_pass2<<<dim3(LL / 1024), blk, 0, stream>>>(mu0, var0, M, S1, S2, MuS, VarS);
        onorm_pass3<<<grd, blk, 0, stream>>>(x, MuS, VarS, out);
    } else {
        onorm_fallback<<<dim3(LL / 1024), dim3(256), 0, stream>>>(x, mu0, var0, out);
    }
}